// HPool_15350213116679
// MI455X (gfx1250) — compile-verified
//
#include <hip/hip_runtime.h>

// Problem constants (from reference): N=64, C=64, H*W=16384, BINS=32
#define CDIM  64
#define HW    16384
#define BINS  32
#define NVEC4 (64 * 64 * HW / 4)   // total float4 elements of x

typedef __attribute__((ext_vector_type(2))) float v2f;
typedef __attribute__((ext_vector_type(8))) float v8f;

// ---- monotone float <-> uint map so we can use integer atomics for min/max ----
__device__ __forceinline__ unsigned int ord_encode(float f) {
  unsigned int u = __float_as_uint(f);
  return (u & 0x80000000u) ? ~u : (u | 0x80000000u);
}
__device__ __forceinline__ float ord_decode(unsigned int u) {
  unsigned int b = (u & 0x80000000u) ? (u ^ 0x80000000u) : ~u;
  return __uint_as_float(b);
}

// ---- hardware tanh if the builtin exists (v_tanh_f32, TRANS op), else a
//      2-trans fallback: tanh(x) = 1 - 2/(exp2(2*log2e*x) + 1)  (saturates to +-1)
__device__ __forceinline__ float fast_tanh(float x) {
#if __has_builtin(__builtin_amdgcn_tanhf)
  return __builtin_amdgcn_tanhf(x);
#elif __has_builtin(__builtin_amdgcn_tanh_f32)
  return __builtin_amdgcn_tanh_f32(x);
#else
  float e = __builtin_amdgcn_exp2f(x * 2.885390081777926f); // 2*log2(e)
  return 1.0f - 2.0f * __builtin_amdgcn_rcpf(e + 1.0f);
#endif
}

__global__ void init_ws_kernel(unsigned int* ws) {
  if (threadIdx.x == 0) {
    ws[0] = 0xFFFFFFFFu;  // encoded +inf-ish (min accumulator)
    ws[1] = 0x00000000u;  // encoded -inf-ish (max accumulator)
  }
}

// ---------------- Pass 1: global min/max over x (streams 256 MB) --------------
__global__ __launch_bounds__(256) void minmax_kernel(const float4* __restrict__ x,
                                                     unsigned int* __restrict__ ws,
                                                     int nvec4) {
  int tid    = blockIdx.x * blockDim.x + threadIdx.x;
  int stride = gridDim.x * blockDim.x;
  float lo =  3.402823466e38f;
  float hi = -3.402823466e38f;
  for (int i = tid; i < nvec4; i += stride) {
    float4 v = x[i];
    lo = fminf(lo, fminf(fminf(v.x, v.y), fminf(v.z, v.w)));
    hi = fmaxf(hi, fmaxf(fmaxf(v.x, v.y), fmaxf(v.z, v.w)));
  }
  // wave32 butterfly reduce
  #pragma unroll
  for (int m = 16; m >= 1; m >>= 1) {
    lo = fminf(lo, __shfl_xor(lo, m, 32));
    hi = fmaxf(hi, __shfl_xor(hi, m, 32));
  }
  __shared__ float slo[8], shi[8];
  int wid = threadIdx.x >> 5;
  if ((threadIdx.x & 31) == 0) { slo[wid] = lo; shi[wid] = hi; }
  __syncthreads();
  if (threadIdx.x == 0) {
    #pragma unroll
    for (int w = 1; w < 8; ++w) { lo = fminf(lo, slo[w]); hi = fmaxf(hi, shi[w]); }
    atomicMin(&ws[0], ord_encode(lo));
    atomicMax(&ws[1], ord_encode(hi));
  }
}

// ------- Pass 2: fused bin -> coeff gather (ds_bpermute) -> tanh -> reduce ----
// One 256-thread block per (n,c) plane; output z[n*C+c] directly (no histogram).
__global__ __launch_bounds__(256) void fused_hist_dot_kernel(
    const float4* __restrict__ x,
    const float*  __restrict__ coeff,
    const unsigned int* __restrict__ ws,
    float* __restrict__ z) {
  const int   nc    = blockIdx.x;          // 0 .. N*C-1
  const int   c     = nc & (CDIM - 1);
  const float xmin  = ord_decode(ws[0]);
  const float xmax  = ord_decode(ws[1]);
  const float scale = (xmax > xmin) ? (float)BINS / (xmax - xmin) : 0.0f;
  const float bias  = -xmin * scale;

  // wave32: lane l caches coeff[c][l]; per-element gather is one ds_bpermute
  const int   lane = threadIdx.x & 31;
  const float cf   = coeff[c * BINS + lane];

  const float4* plane = x + (size_t)nc * (HW / 4);
  float acc = 0.0f;

  #pragma unroll 4
  for (int i = threadIdx.x; i < HW / 4; i += 256) {
    __builtin_prefetch(plane + i + 512, 0, 3);   // global_prefetch_b8 ahead of stream
    float4 v = plane[i];
    float vv[4] = {v.x, v.y, v.z, v.w};
    #pragma unroll
    for (int j = 0; j < 4; ++j) {
      float f   = vv[j];
      int   idx = (int)__builtin_fmaf(f, scale, bias);   // floor for vals in [0,32]
      idx = idx < 0 ? 0 : (idx > BINS - 1 ? BINS - 1 : idx);
      float w = __shfl(cf, idx, 32);                     // ds_bpermute_b32
      acc = __builtin_fmaf(fast_tanh(f), w, acc);
    }
  }

  // ---- exact f32 wave reduction via V_WMMA_F32_16X16X4_F32 ----
  // A(16x4): each lane contributes {acc, 0} in its K-slots; B = all ones.
  // D rows hold exact f32 row-sums; per-lane sum of the 8 D VGPRs gives the
  // sum of 8 row-sums; one shfl_xor(16) completes the 32-lane total.
  v2f a; a.x = acc;  a.y = 0.0f;
  v2f b; b.x = 1.0f; b.y = 1.0f;
  v8f cz = {};
  v8f d = __builtin_amdgcn_wmma_f32_16x16x4_f32(
      /*neg_a=*/false, a, /*neg_b=*/false, b,
      /*c_mod=*/(short)0, cz, /*reuse_a=*/false, /*reuse_b=*/false);
  float s = 0.0f;
  #pragma unroll
  for (int j = 0; j < 8; ++j) s += d[j];
  s += __shfl_xor(s, 16, 32);          // wave total, uniform across lanes

  __shared__ float swsum[8];
  int wid = threadIdx.x >> 5;
  if ((threadIdx.x & 31) == 0) swsum[wid] = s;
  __syncthreads();
  if (threadIdx.x == 0) {
    float t = 0.0f;
    #pragma unroll
    for (int w = 0; w < 8; ++w) t += swsum[w];
    z[nc] = t;
  }
}

extern "C" void kernel_launch(void* const* d_in, const int* in_sizes, int n_in,
                              void* d_out, int out_size, void* d_ws, size_t ws_size,
                              hipStream_t stream) {
  const float4* x     = (const float4*)d_in[0];           // (64,64,128,128) f32
  const float*  coeff = (const float*)d_in[1];            // (64,32) f32
  float*        z     = (float*)d_out;                    // (64,64) f32
  unsigned int* ws    = (unsigned int*)d_ws;              // [0]=enc-min, [1]=enc-max

  init_ws_kernel<<<1, 32, 0, stream>>>(ws);
  minmax_kernel<<<8192, 256, 0, stream>>>(x, ws, NVEC4);
  fused_hist_dot_kernel<<<64 * CDIM, 256, 0, stream>>>(x, coeff, ws, z);
}